// GMM_57750130262436
// MI455X (gfx1250) — compile-verified
//
#include <hip/hip_runtime.h>
#include <math.h>

typedef __attribute__((ext_vector_type(2))) float v2f;
typedef __attribute__((ext_vector_type(8))) float v8f;

#define NB   2048
#define DIMS 32
#define NC   32
#define NS   528          // 32*33/2
#define MM   100
#define C0   (-29.40603306254953f)  // -16*log(2*pi)
#define LOGM 4.605170185988091f     // log(100)

// workspace layout, in floats
#define WS_ZC    0        // 2048*32 = 65536
#define WS_LOGPI 65536    // 528
#define WS_G     66064    // 1024
#define WS_P     67088    // 52800
// total 119888 floats ~= 480 KB

// ---------------- kernel 1: softmax(pi), copy mu, Gram matrix ----------------
__global__ void GMM_setup_kernel(const float* __restrict__ pi,
                                 const float* __restrict__ mu,
                                 float* __restrict__ out,
                                 float* __restrict__ ws) {
  __shared__ float red[1024];
  const int tid = threadIdx.x;                 // 1024 threads, 1 block

  float v = (tid < NS) ? pi[tid] : -INFINITY;
  red[tid] = v;
  __syncthreads();
  for (int off = 512; off > 0; off >>= 1) {
    if (tid < off) red[tid] = fmaxf(red[tid], red[tid + off]);
    __syncthreads();
  }
  const float pmax = red[0];
  __syncthreads();
  const float e = (tid < NS) ? __expf(v - pmax) : 0.0f;
  red[tid] = e;
  __syncthreads();
  for (int off = 512; off > 0; off >>= 1) {
    if (tid < off) red[tid] += red[tid + off];
    __syncthreads();
  }
  const float psum = red[0];
  if (tid < NS) {
    const float soft = e / psum;
    out[tid] = soft;                           // pi_soft -> out[0..527]
    ws[WS_LOGPI + tid] = __logf(soft + 1e-30f);
  }
  out[NS + tid] = mu[tid];                     // mu -> out[528..1551]

  // Gram G[i][j] = sum_d mu[d,i]*mu[d,j]
  const int i = tid >> 5, j = tid & 31;
  float g = 0.0f;
  for (int d = 0; d < DIMS; ++d) g += mu[d * NC + i] * mu[d * NC + j];
  ws[WS_G + tid] = g;
}

// ---------------- kernel 2: zc = z @ mu via V_WMMA_F32_16X16X4_F32 ----------------
// One wave per block; 16 rows x 32 cols output slab; K=32 in chunks of 4.
__global__ void GMM_zc_wmma_kernel(const float* __restrict__ z,
                                   const float* __restrict__ mu,
                                   float* __restrict__ ws) {
  const int lane = threadIdx.x;        // 0..31, EXEC all-ones (no divergence)
  const int half = lane >> 4;
  const int l    = lane & 15;
  const int rowbase = blockIdx.x * 16;

  v8f acc0 = {};
  v8f acc1 = {};
  #pragma unroll
  for (int k0 = 0; k0 < DIMS; k0 += 4) {
    const int ka = k0 + 2 * half;
    // A 16x4 f32: lanes 0-15 -> M=l with K={k0,k0+1}; lanes 16-31 -> K={k0+2,k0+3}
    v2f a;
    a.x = z[(rowbase + l) * DIMS + ka];
    a.y = z[(rowbase + l) * DIMS + ka + 1];
    // B 4x16 f32: N across lanes, K split across halves/VGPRs (mirror of A)
    v2f b0, b1;
    b0.x = mu[ka * NC + l];
    b0.y = mu[(ka + 1) * NC + l];
    b1.x = mu[ka * NC + 16 + l];
    b1.y = mu[(ka + 1) * NC + 16 + l];
    acc0 = __builtin_amdgcn_wmma_f32_16x16x4_f32(false, a, false, b0,
                                                 (short)0, acc0, false, false);
    acc1 = __builtin_amdgcn_wmma_f32_16x16x4_f32(false, a, false, b1,
                                                 (short)0, acc1, false, false);
  }
  // D layout: VGPR r -> M=r (lanes 0-15), M=8+r (lanes 16-31); N = lane%16
  #pragma unroll
  for (int r = 0; r < 8; ++r) {
    const int row = rowbase + r + 8 * half;
    ws[WS_ZC + row * NC + l]      = acc0[r];
    ws[WS_ZC + row * NC + 16 + l] = acc1[r];
  }
}

// ---------------- kernel 3: P[s,m] table ----------------
__global__ void GMM_p_kernel(float* __restrict__ ws) {
  const int idx = blockIdx.x * blockDim.x + threadIdx.x;
  if (idx >= NS * MM) return;
  const int s = idx / MM;
  const int m = idx - s * MM;
  // triu index map: s -> (a, c), a<=c  (np.nonzero(np.triu(ones)) order)
  int rem = s, a = 0;
  while (rem >= NC - a) { rem -= NC - a; ++a; }
  const int c = a + rem;
  const float w = (float)m * (1.0f / MM);
  const float u = 1.0f - w;
  const float* G = ws + WS_G;
  const float quad = w * w * G[a * NC + a] + 2.0f * w * u * G[a * NC + c]
                   + u * u * G[c * NC + c];
  ws[WS_P + idx] = ws[WS_LOGPI + s] - 0.5f * quad;
}

// ---------------- kernel 4: fused logsumexp over (s,m) ----------------
#define BB 8   // batch rows per block
__global__ void GMM_lse_kernel(const float* __restrict__ z,
                               const float* __restrict__ ws,
                               float* __restrict__ out) {
  __shared__ float lzc[BB][NC];
  __shared__ float lz2[BB];
  __shared__ unsigned char la[NS], lc[NS];
  __shared__ float redAll[BB][256];
  __shared__ float gmax[BB];

  const int tid = threadIdx.x;                 // 256 threads
  const int b0  = blockIdx.x * BB;

  {
    const int r = tid >> 5, d = tid & 31;      // 8 rows x 32 dims
    lzc[r][d] = ws[WS_ZC + (b0 + r) * NC + d];
  }
  if (tid < BB) {
    float s2 = 0.0f;
    for (int d = 0; d < DIMS; ++d) { const float t = z[(b0 + tid) * DIMS + d]; s2 += t * t; }
    lz2[tid] = s2;
  }
  for (int s = tid; s < NS; s += 256) {
    int rem = s, a = 0;
    while (rem >= NC - a) { rem -= NC - a; ++a; }
    la[s] = (unsigned char)a;
    lc[s] = (unsigned char)(a + rem);
  }
  __syncthreads();

  const float* P = ws + WS_P;

  // ---- pass 1: max ----
  float mx[BB];
  #pragma unroll
  for (int i = 0; i < BB; ++i) mx[i] = -INFINITY;
  for (int s = tid; s < NS; s += 256) {
    const int a = la[s], c = lc[s];
    float dr[BB], er[BB];
    #pragma unroll
    for (int i = 0; i < BB; ++i) {
      const float za = lzc[i][a], zb = lzc[i][c];
      dr[i] = za - zb; er[i] = zb;
    }
    const float* Ps = P + s * MM;
    for (int m = 0; m < MM; ++m) {
      const float p = Ps[m];
      const float w = (float)m * (1.0f / MM);
      #pragma unroll
      for (int i = 0; i < BB; ++i) mx[i] = fmaxf(mx[i], fmaf(w, dr[i], p + er[i]));
    }
  }
  #pragma unroll
  for (int i = 0; i < BB; ++i) redAll[i][tid] = mx[i];
  __syncthreads();
  if (tid < BB) {
    float g = -INFINITY;
    for (int j = 0; j < 256; ++j) g = fmaxf(g, redAll[tid][j]);
    gmax[tid] = g;
  }
  __syncthreads();

  float gm[BB], sm[BB];
  #pragma unroll
  for (int i = 0; i < BB; ++i) { gm[i] = gmax[i]; sm[i] = 0.0f; }

  // ---- pass 2: sum of exp(arg - max) ----
  for (int s = tid; s < NS; s += 256) {
    const int a = la[s], c = lc[s];
    float dr[BB], er[BB];
    #pragma unroll
    for (int i = 0; i < BB; ++i) {
      const float za = lzc[i][a], zb = lzc[i][c];
      dr[i] = za - zb; er[i] = zb - gm[i];
    }
    const float* Ps = P + s * MM;
    for (int m = 0; m < MM; ++m) {
      const float p = Ps[m];
      const float w = (float)m * (1.0f / MM);
      #pragma unroll
      for (int i = 0; i < BB; ++i) sm[i] += __expf(fmaf(w, dr[i], p + er[i]));
    }
  }
  #pragma unroll
  for (int i = 0; i < BB; ++i) redAll[i][tid] = sm[i];
  __syncthreads();
  if (tid < BB) {
    float t = 0.0f;
    for (int j = 0; j < 256; ++j) t += redAll[tid][j];
    out[NS + NC * DIMS + b0 + tid] =
        C0 - 0.5f * lz2[tid] + gmax[tid] + __logf(t) - LOGM;
  }
}

extern "C" void kernel_launch(void* const* d_in, const int* in_sizes, int n_in,
                              void* d_out, int out_size, void* d_ws, size_t ws_size,
                              hipStream_t stream) {
  const float* z  = (const float*)d_in[0];   // [2048,32]
  const float* pi = (const float*)d_in[1];   // [1,528]
  const float* mu = (const float*)d_in[2];   // [32,32]
  float* out = (float*)d_out;                // 528 + 1024 + 2048 = 3600 floats
  float* ws  = (float*)d_ws;

  GMM_setup_kernel<<<1, 1024, 0, stream>>>(pi, mu, out, ws);
  GMM_zc_wmma_kernel<<<NB / 16, 32, 0, stream>>>(z, mu, ws);
  GMM_p_kernel<<<(NS * MM + 255) / 256, 256, 0, stream>>>(ws);
  GMM_lse_kernel<<<NB / BB, 256, 0, stream>>>(z, ws, out);
}